// DendriticBranchLayer_63428077027580
// MI455X (gfx1250) — compile-verified
//
#include <hip/hip_runtime.h>

#define BATCH 4096
#define D_EXC 8192
#define D_INH 2048
#define D_OUT 4096
#define K_EXC 32
#define K_INH 16

// ---------------------------------------------------------------------------
// Helpers
// ---------------------------------------------------------------------------
__device__ __forceinline__ unsigned long long pack_key(float v, int idx) {
    // Order-preserving float->u32 map, then pack with ~idx so that on value
    // ties the SMALLER index wins the max (matches jax.lax.top_k ordering).
    unsigned u = __float_as_uint(v);
    u = (u & 0x80000000u) ? ~u : (u | 0x80000000u);
    return ((unsigned long long)u << 32) | (unsigned)(0xFFFFFFFFu - (unsigned)idx);
}

__device__ __forceinline__ float unpack_val(unsigned long long key) {
    unsigned m = (unsigned)(key >> 32);
    unsigned bits = (m & 0x80000000u) ? (m ^ 0x80000000u) : ~m;
    return __uint_as_float(bits);
}

__device__ __forceinline__ unsigned long long shfl_xor_u64(unsigned long long x, int m) {
    unsigned lo = (unsigned)x, hi = (unsigned)(x >> 32);
    lo = (unsigned)__shfl_xor((int)lo, m, 32);   // wave32-width shuffles
    hi = (unsigned)__shfl_xor((int)hi, m, 32);
    return ((unsigned long long)hi << 32) | lo;
}

// ---------------------------------------------------------------------------
// Kernel 1: per-row top-K compaction.
//   Row elements live in registers (NPT per thread, 256 threads).
//   K iterations of block-wide argmax; winner's weight = exp(value).
//   Output table is transposed [K][D_OUT] and PACKED: (f32 weight, u32 idx)
//   in one 8-byte entry so the main kernel does a single b64 load per MAC.
// ---------------------------------------------------------------------------
template <int NPT, int K>
__global__ __launch_bounds__(256)
void topk_compact_kernel(const float* __restrict__ pre_w, int d_in, int d_out,
                         uint2* __restrict__ pT) {
    const int o   = blockIdx.x;
    const int tid = threadIdx.x;
    const float* row = pre_w + (size_t)o * d_in;

    float v[NPT];
#pragma unroll
    for (int j = 0; j < NPT; ++j) v[j] = row[j * 256 + tid];

    __shared__ unsigned long long wavemax[8];
    __shared__ unsigned long long winner;

    for (int kk = 0; kk < K; ++kk) {
        // per-thread best
        unsigned long long best = pack_key(v[0], tid);
#pragma unroll
        for (int j = 1; j < NPT; ++j) {
            unsigned long long c = pack_key(v[j], j * 256 + tid);
            if (c > best) best = c;
        }
        // wave32 reduction
#pragma unroll
        for (int m = 16; m > 0; m >>= 1) {
            unsigned long long other = shfl_xor_u64(best, m);
            if (other > best) best = other;
        }
        if ((tid & 31) == 0) wavemax[tid >> 5] = best;
        __syncthreads();
        if (tid == 0) {
            unsigned long long b = wavemax[0];
#pragma unroll
            for (int w = 1; w < 8; ++w)
                if (wavemax[w] > b) b = wavemax[w];
            winner = b;
        }
        __syncthreads();
        const unsigned long long b = winner;
        const int widx = (int)(0xFFFFFFFFu - (unsigned)b);
        // owner knocks the winner out for subsequent iterations
        if ((widx & 255) == tid) v[widx >> 8] = -__builtin_inff();
        if (tid == 0) {
            pT[(size_t)kk * d_out + o] =
                make_uint2(__float_as_uint(expf(unpack_val(b))), (unsigned)widx);
        }
        // NOTE: no barrier needed here. All threads read `winner` before they
        // can reach the next iteration's first __syncthreads(), and thread 0
        // only rewrites `winner` after passing that barrier.
    }
}

// ---------------------------------------------------------------------------
// Kernel 2: main sparse gather-FMA.
//   One block per batch row. x rows staged to LDS via CDNA5 async
//   global->LDS DMA (ASYNCcnt path), then 256 threads each produce 16
//   outputs. Per MAC: one coalesced global_load_b64 (packed w,idx), one
//   ds_load gather, one v_fma -> 3 dynamic instructions per MAC.
// ---------------------------------------------------------------------------
#define OPT 16  // outputs per thread: 256 * 16 = 4096 = D_OUT

__global__ __launch_bounds__(256)
void dendritic_main_kernel(const float* __restrict__ x_exc,
                           const float* __restrict__ x_inh,
                           const uint2* __restrict__ peT,
                           const uint2* __restrict__ piT,
                           float* __restrict__ out) {
    __shared__ __align__(16) float lx_exc[D_EXC];   // 32 KB
    __shared__ __align__(16) float lx_inh[D_INH];   //  8 KB

    const int tid = threadIdx.x;
    const int b   = blockIdx.x;

    // ---- async stage x_exc[b,:] and x_inh[b,:] into LDS -------------------
    {
        unsigned long long gsrc =
            (unsigned long long)(x_exc + (size_t)b * D_EXC) + (unsigned long long)tid * 16ull;
        // low 32 bits of a flat shared address == LDS byte offset (aperture map)
        unsigned ldst = (unsigned)(size_t)(&lx_exc[0]) + (unsigned)tid * 16u;
#pragma unroll
        for (int it = 0; it < (D_EXC * 4) / (256 * 16); ++it) {  // 8 sweeps
            asm volatile("global_load_async_to_lds_b128 %0, %1, off"
                         :: "v"(ldst + (unsigned)it * 4096u),
                            "v"(gsrc + (unsigned long long)it * 4096ull)
                         : "memory");
        }
        unsigned long long gsrc2 =
            (unsigned long long)(x_inh + (size_t)b * D_INH) + (unsigned long long)tid * 16ull;
        unsigned ldst2 = (unsigned)(size_t)(&lx_inh[0]) + (unsigned)tid * 16u;
#pragma unroll
        for (int it = 0; it < (D_INH * 4) / (256 * 16); ++it) {  // 2 sweeps
            asm volatile("global_load_async_to_lds_b128 %0, %1, off"
                         :: "v"(ldst2 + (unsigned)it * 4096u),
                            "v"(gsrc2 + (unsigned long long)it * 4096ull)
                         : "memory");
        }
        asm volatile("s_wait_asynccnt 0" ::: "memory");
    }
    __syncthreads();

    // ---- excitatory accumulation -----------------------------------------
    float e[OPT];
#pragma unroll
    for (int j = 0; j < OPT; ++j) e[j] = 0.0f;
    for (int k = 0; k < K_EXC; ++k) {
#pragma unroll
        for (int j = 0; j < OPT; ++j) {
            const int o = tid + j * 256;
            const uint2 t = peT[(size_t)k * D_OUT + o];       // one b64 load
            e[j] = fmaf(__uint_as_float(t.x), lx_exc[t.y], e[j]);
        }
    }

    // ---- inhibitory accumulation -----------------------------------------
    float ia[OPT];
#pragma unroll
    for (int j = 0; j < OPT; ++j) ia[j] = 0.0f;
    for (int k = 0; k < K_INH; ++k) {
#pragma unroll
        for (int j = 0; j < OPT; ++j) {
            const int o = tid + j * 256;
            const uint2 t = piT[(size_t)k * D_OUT + o];       // one b64 load
            ia[j] = fmaf(__uint_as_float(t.x), lx_inh[t.y], ia[j]);
        }
    }

    // ---- shunting inhibition + coalesced store ---------------------------
#pragma unroll
    for (int j = 0; j < OPT; ++j) {
        const int o = tid + j * 256;
        out[(size_t)b * D_OUT + o] = e[j] / (1.0f + ia[j]);
    }
}

// ---------------------------------------------------------------------------
// Launch
// ---------------------------------------------------------------------------
extern "C" void kernel_launch(void* const* d_in, const int* in_sizes, int n_in,
                              void* d_out, int out_size, void* d_ws, size_t ws_size,
                              hipStream_t stream) {
    const float* x_exc  = (const float*)d_in[0];  // [4096, 8192]
    const float* x_inh  = (const float*)d_in[1];  // [4096, 2048]
    const float* pw_exc = (const float*)d_in[2];  // [4096, 8192]
    const float* pw_inh = (const float*)d_in[3];  // [4096, 2048]
    float*       out    = (float*)d_out;          // [4096, 4096]

    // workspace carve: packed tables, 8-byte entries
    char* ws = (char*)d_ws;
    uint2* peT = (uint2*)ws;                                       // 32*4096*8 = 1 MB
    uint2* piT = (uint2*)(ws + (size_t)K_EXC * D_OUT * 8);         // 16*4096*8 = 512 KB

    topk_compact_kernel<D_EXC / 256, K_EXC><<<D_OUT, 256, 0, stream>>>(
        pw_exc, D_EXC, D_OUT, peT);
    topk_compact_kernel<D_INH / 256, K_INH><<<D_OUT, 256, 0, stream>>>(
        pw_inh, D_INH, D_OUT, piT);
    dendritic_main_kernel<<<BATCH, 256, 0, stream>>>(
        x_exc, x_inh, peT, piT, out);
}